// NTK_PINN_NS2D_163208757354
// MI455X (gfx1250) — compile-verified
//
#include <hip/hip_runtime.h>
#include <math.h>

typedef __attribute__((ext_vector_type(2))) float v2f;
typedef __attribute__((ext_vector_type(8))) float v8f;

#define NN 4096
#define CSPLIT 4
#define WAVES_PER_BLOCK 8
#define TILES_PER_BLOCK ((NN / CSPLIT) / 16)                 // 64 column tiles per block
#define TILES_PER_WAVE (TILES_PER_BLOCK / WAVES_PER_BLOCK)   // 8 tiles per wave

#define PI_F      3.14159265358979323846f
#define INV_2PI_F 0.15915494309189535f
#define CLIP_LO  (-1.0f + 1e-7f)
#define CLIP_HI  ( 1.0f - 1e-7f)

// One relu_ntk layer applied to a single element, given row/col inv-sqrt scalars.
__device__ __forceinline__ float ntk_layer(float k, float rsr, float rsc) {
    float c = k * rsr * rsc;
    c = fminf(fmaxf(c, CLIP_LO), CLIP_HI);
    float th = acosf(c);
    float s  = sqrtf(1.0f - c * c);
    return INV_2PI_F * (c * (PI_F - th) + s);   // sigma_w=1, sigma_b=0
}

// ---------------------------------------------------------------------------
// Kernel 1: diagonal recurrence.  rs[l][i] = 1/sqrt(diag_l[i] + eps), l=0..3.
// diag_0[i] = (1/3) * dot(X[i], X_train[i]);  diag_{l+1}[i] = relu_ntk(diag)_ii.
// ---------------------------------------------------------------------------
__global__ void ntk_diag_kernel(const float* __restrict__ x,
                                const float* __restrict__ y,
                                const float* __restrict__ t,
                                const float* __restrict__ Xt,
                                float* __restrict__ rs) {
    int i = blockIdx.x * blockDim.x + threadIdx.x;
    if (i >= NN) return;
    float d = (x[i] * Xt[3 * i + 0] + y[i] * Xt[3 * i + 1] + t[i] * Xt[3 * i + 2])
              * (1.0f / 3.0f);
#pragma unroll
    for (int l = 0; l < 4; ++l) {
        float inv = 1.0f / sqrtf(d + 1e-12f);
        rs[l * NN + i] = inv;
        float c = d * inv * inv;
        c = fminf(fmaxf(c, CLIP_LO), CLIP_HI);
        float th = acosf(c);
        d = INV_2PI_F * (c * (PI_F - th) + sqrtf(1.0f - c * c));
    }
}

// ---------------------------------------------------------------------------
// Kernel 2: fused K0-WMMA + 4 NTK layers + K@alpha partial reduction.
// Block = (row strip of 16) x (1024-column chunk), 8 waves, wave32.
// ---------------------------------------------------------------------------
__global__ __launch_bounds__(256) void ntk_tile_kernel(
    const float* __restrict__ x, const float* __restrict__ y,
    const float* __restrict__ t, const float* __restrict__ Xt,
    const float* __restrict__ alpha, const float* __restrict__ rs,
    float* __restrict__ part) {
    const int strip = blockIdx.x;           // 0..255 : rows [strip*16, strip*16+16)
    const int cs    = blockIdx.y;           // 0..CSPLIT-1 : column chunk
    const int i0    = strip * 16;
    const int lane  = threadIdx.x & 31;
    const int wave  = threadIdx.x >> 5;
    const int half  = lane >> 4;            // 0: K=0,1 slice / rows 0..7 of C
    const int ln    = lane & 15;            // M (for A) or N (for B/C)

    // A tile: 16x4 fp32 (K-dim padded 3->4).  Lane layout per ISA:
    // lanes 0-15: M=ln, VGPR0=K0, VGPR1=K1 ; lanes 16-31: M=ln, VGPR0=K2, VGPR1=K3.
    const int irow = i0 + ln;
    v2f a;
    a.x = half ? t[irow] : x[irow];
    a.y = half ? 0.0f    : y[irow];

    // Row inv-sqrt scalars for this lane's 8 accumulator rows (i0 + half*8 + v),
    // invariant across the column loop.
    float rsr[4][8];
#pragma unroll
    for (int l = 0; l < 4; ++l)
#pragma unroll
        for (int v = 0; v < 8; ++v)
            rsr[l][v] = rs[l * NN + i0 + half * 8 + v];

    float accu[8], accv[8], accp[8];
#pragma unroll
    for (int v = 0; v < 8; ++v) { accu[v] = 0.0f; accv[v] = 0.0f; accp[v] = 0.0f; }

    for (int ti = 0; ti < TILES_PER_WAVE; ++ti) {
        const int jt = cs * TILES_PER_BLOCK + ti * WAVES_PER_BLOCK + wave;
        const int j  = jt * 16 + ln;        // this lane's column index

        // B tile: 4x16 fp32 of X_train^T (mirror of A layout).
        v2f b;
        b.x = half ? Xt[3 * j + 2] : Xt[3 * j + 0];
        b.y = half ? 0.0f          : Xt[3 * j + 1];

        // K0 tile = A x B  (scaled by 1/3 afterwards)
        v8f c = {};
        c = __builtin_amdgcn_wmma_f32_16x16x4_f32(
                /*neg_a=*/false, a, /*neg_b=*/false, b,
                /*c_mod=*/(short)0, c, /*reuse_a=*/false, /*reuse_b=*/false);

        float k[8];
#pragma unroll
        for (int v = 0; v < 8; ++v) k[v] = c[v] * (1.0f / 3.0f);

        // 4 fused relu_ntk layers, all in registers.
#pragma unroll
        for (int l = 0; l < 4; ++l) {
            const float rsc = rs[l * NN + j];
#pragma unroll
            for (int v = 0; v < 8; ++v) k[v] = ntk_layer(k[v], rsr[l][v], rsc);
        }

        // Fold in alpha[j] (K_star @ alpha partial sums).
        const float a0 = alpha[3 * j + 0];
        const float a1 = alpha[3 * j + 1];
        const float a2 = alpha[3 * j + 2];
#pragma unroll
        for (int v = 0; v < 8; ++v) {
            accu[v] = fmaf(k[v], a0, accu[v]);
            accv[v] = fmaf(k[v], a1, accv[v]);
            accp[v] = fmaf(k[v], a2, accp[v]);
        }
    }

    // Deterministic butterfly reduction over the 16 lanes of each half
    // (each half holds disjoint rows: half*8 + v).
#pragma unroll
    for (int off = 1; off < 16; off <<= 1) {
#pragma unroll
        for (int v = 0; v < 8; ++v) {
            accu[v] += __shfl_xor(accu[v], off, 32);
            accv[v] += __shfl_xor(accv[v], off, 32);
            accp[v] += __shfl_xor(accp[v], off, 32);
        }
    }

    __shared__ float s_part[WAVES_PER_BLOCK][16][3];
    if (ln == 0) {
#pragma unroll
        for (int v = 0; v < 8; ++v) {
            const int r = half * 8 + v;
            s_part[wave][r][0] = accu[v];
            s_part[wave][r][1] = accv[v];
            s_part[wave][r][2] = accp[v];
        }
    }
    __syncthreads();

    // Fixed-order cross-wave sum -> per-(strip,chunk) partials in workspace.
    if (threadIdx.x < 48) {
        const int r    = threadIdx.x & 15;
        const int comp = threadIdx.x >> 4;
        float s = 0.0f;
#pragma unroll
        for (int w = 0; w < WAVES_PER_BLOCK; ++w) s += s_part[w][r][comp];
        part[((strip * CSPLIT + cs) * 3 + comp) * 16 + r] = s;
    }
}

// ---------------------------------------------------------------------------
// Kernel 3: sum the CSPLIT column-chunk partials -> out (u[0:N] | v[0:N] | p[0:N]).
// ---------------------------------------------------------------------------
__global__ void ntk_finalize_kernel(const float* __restrict__ part,
                                    float* __restrict__ out) {
    int idx = blockIdx.x * blockDim.x + threadIdx.x;
    if (idx >= 3 * NN) return;
    const int comp  = idx / NN;
    const int i     = idx - comp * NN;
    const int strip = i >> 4;
    const int r     = i & 15;
    float s = 0.0f;
#pragma unroll
    for (int csx = 0; csx < CSPLIT; ++csx)
        s += part[((strip * CSPLIT + csx) * 3 + comp) * 16 + r];
    out[idx] = s;
}

extern "C" void kernel_launch(void* const* d_in, const int* in_sizes, int n_in,
                              void* d_out, int out_size, void* d_ws, size_t ws_size,
                              hipStream_t stream) {
    (void)in_sizes; (void)n_in; (void)out_size; (void)ws_size;
    const float* x     = (const float*)d_in[0];
    const float* y     = (const float*)d_in[1];
    const float* t     = (const float*)d_in[2];
    const float* Xt    = (const float*)d_in[3];
    const float* alpha = (const float*)d_in[4];
    float* out = (float*)d_out;

    float* rs   = (float*)d_ws;                                   // 4*NN floats (64 KB)
    float* part = (float*)((char*)d_ws + 4 * NN * sizeof(float)); // 256*CSPLIT*48 floats (192 KB)

    ntk_diag_kernel<<<NN / 256, 256, 0, stream>>>(x, y, t, Xt, rs);

    dim3 grid(NN / 16, CSPLIT);   // 256 row strips x 4 column chunks
    ntk_tile_kernel<<<grid, 256, 0, stream>>>(x, y, t, Xt, alpha, rs, part);

    ntk_finalize_kernel<<<(3 * NN + 255) / 256, 256, 0, stream>>>(part, out);
}